// BillehColumn_4861902979703
// MI455X (gfx1250) — compile-verified
//
#include <hip/hip_runtime.h>

#define NN 100000              // neurons
#define RR 4                   // receptors
#define NR (NN * RR)           // 400000 slots
#define EE 2000000             // synapses (multiple of 4)
#define TT 50                  // timesteps
#define ZWORDS 3125            // NN/32 exactly
#define ZWORDS_PAD 3136        // padded to multiple of 4 words (b128 staging)

#if defined(__HIP_DEVICE_COMPILE__) && __has_builtin(__builtin_amdgcn_global_load_async_to_lds_b128)
#define HAVE_ASYNC_LDS 1
#else
#define HAVE_ASYNC_LDS 0
#endif

typedef int v4i __attribute__((ext_vector_type(4)));
typedef __attribute__((address_space(1))) v4i* g_v4i_ptr;   // global
typedef __attribute__((address_space(3))) v4i* l_v4i_ptr;   // LDS

__device__ __forceinline__ unsigned ballot32(bool p) {
#if defined(__HIP_DEVICE_COMPILE__) && __has_builtin(__builtin_amdgcn_ballot_w32)
  return __builtin_amdgcn_ballot_w32(p);
#else
  return (unsigned)__ballot((int)p);
#endif
}

// ---------------------------------------------------------------------------
// Init: zero all recurrent state, copy v0, zero spike bitmask.
// Runs once per kernel_launch so graph replays are deterministic.
// ---------------------------------------------------------------------------
__global__ void __launch_bounds__(256) init_state_kernel(
    const float* __restrict__ v0,
    float* __restrict__ z, float* __restrict__ v, float* __restrict__ r,
    float* __restrict__ asc, float* __restrict__ psc,
    float* __restrict__ psc_rise, float* __restrict__ rec_in,
    unsigned* __restrict__ z_bits)
{
  int i = blockIdx.x * blockDim.x + threadIdx.x;
  if (i < NR) { psc[i] = 0.0f; psc_rise[i] = 0.0f; rec_in[i] = 0.0f; }
  if (i < 2 * NN) { asc[i] = 0.0f; }
  if (i < NN) { z[i] = 0.0f; v[i] = v0[i]; r[i] = 0.0f; }
  if (i < ZWORDS_PAD) { z_bits[i] = 0u; }
}

// ---------------------------------------------------------------------------
// Prep: fuse post*4 + receptor into one slot index (amortized over T steps).
// ---------------------------------------------------------------------------
__global__ void __launch_bounds__(256) prep_slots_kernel(
    const int* __restrict__ post_idx, const int* __restrict__ rcpt_idx,
    int* __restrict__ slot)
{
  int e = blockIdx.x * blockDim.x + threadIdx.x;
  if (e < EE) slot[e] = (post_idx[e] << 2) + rcpt_idx[e];
}

// ---------------------------------------------------------------------------
// Scatter: rec_in[slot] += w for every synapse whose pre neuron spiked.
// Spike bitmask (12.5 KB) is staged into LDS with CDNA5 async global->LDS
// loads; pre_idx is streamed as b128 (4 synapses per load); each synapse
// costs one LDS probe + (if pre spiked) two gathers and one f32 L2 atomic.
// ---------------------------------------------------------------------------
__global__ void __launch_bounds__(256) scatter_kernel(
    const int*   __restrict__ pre_idx,
    const int*   __restrict__ slot,
    const float* __restrict__ w_rec,
    const unsigned* __restrict__ z_bits,
    float* __restrict__ rec_in)
{
  __shared__ unsigned zb[ZWORDS_PAD];

#if HAVE_ASYNC_LDS
  for (int w = (int)threadIdx.x * 4; w < ZWORDS_PAD; w += 256 * 4) {
    __builtin_amdgcn_global_load_async_to_lds_b128(
        (g_v4i_ptr)(z_bits + w),
        (l_v4i_ptr)(&zb[w]), 0, 0);
  }
#if defined(__HIP_DEVICE_COMPILE__) && __has_builtin(__builtin_amdgcn_s_wait_asynccnt)
  __builtin_amdgcn_s_wait_asynccnt(0);
#else
  asm volatile("s_wait_asynccnt 0" ::: "memory");
#endif
  __syncthreads();
#else
  for (int w = (int)threadIdx.x; w < ZWORDS_PAD; w += 256) zb[w] = z_bits[w];
  __syncthreads();
#endif

  const v4i* __restrict__ pre4 = (const v4i*)pre_idx;
  const int nquads = EE / 4;
  const int stride = (int)gridDim.x * 256;
  for (int q = (int)blockIdx.x * 256 + (int)threadIdx.x; q < nquads; q += stride) {
    const v4i p = pre4[q];
    const int base = q * 4;
#pragma unroll
    for (int k = 0; k < 4; ++k) {
      const int pre = p[k];
      if (zb[pre >> 5] & (1u << (pre & 31))) {
        atomicAdd(rec_in + slot[base + k], w_rec[base + k]);  // rec_z == z in {0,1}
      }
    }
  }
}

// ---------------------------------------------------------------------------
// Per-neuron GLIF3 update for one timestep. Consumes rec_in and zeroes it
// for the next step; emits spikes (float) and the spike bitmask via ballot.
// ---------------------------------------------------------------------------
__global__ void __launch_bounds__(256) neuron_kernel(
    const float* __restrict__ x_t,        // x_ext + t*NR
    const float* __restrict__ x_next,     // x_ext + (t+1)*NR, or null
    const float* __restrict__ v_th,
    const float* __restrict__ v_reset,
    const float* __restrict__ t_ref,
    const float* __restrict__ decay,
    const float* __restrict__ cur_f,
    const float* __restrict__ e_l,
    const float* __restrict__ asc_amps,
    const float* __restrict__ asc_decay,
    const float* __restrict__ syn_decay,  // [4]
    const float* __restrict__ psc_init,   // [4]
    float* __restrict__ z, float* __restrict__ v, float* __restrict__ r,
    float* __restrict__ asc, float* __restrict__ psc,
    float* __restrict__ psc_rise, float* __restrict__ rec_in,
    unsigned* __restrict__ z_bits,
    float* __restrict__ out_t)            // out + t*NN
{
  const int i = (int)blockIdx.x * 256 + (int)threadIdx.x;
  if (i >= NN) return;

  const float4 sd = *(const float4*)syn_decay;
  const float4 pi = *(const float4*)psc_init;

  const float4 ri4 = *(const float4*)(rec_in + 4 * i);
  const float4 xt4 = *(const float4*)(x_t + 4 * i);
  const float4 pr4 = *(const float4*)(psc_rise + 4 * i);
  const float4 pc4 = *(const float4*)(psc + 4 * i);
  if (x_next) __builtin_prefetch(x_next + 4 * i, 0, 3);  // warm L2 for t+1

  // inputs = rec_in + x_t ; double-exponential filter (DT = 1)
  float4 npr, npc;
  npr.x = fmaf(pr4.x, sd.x, (ri4.x + xt4.x) * pi.x);
  npr.y = fmaf(pr4.y, sd.y, (ri4.y + xt4.y) * pi.y);
  npr.z = fmaf(pr4.z, sd.z, (ri4.z + xt4.z) * pi.z);
  npr.w = fmaf(pr4.w, sd.w, (ri4.w + xt4.w) * pi.w);
  npc.x = fmaf(pc4.x, sd.x, sd.x * pr4.x);
  npc.y = fmaf(pc4.y, sd.y, sd.y * pr4.y);
  npc.z = fmaf(pc4.z, sd.z, sd.z * pr4.z);
  npc.w = fmaf(pc4.w, sd.w, sd.w * pr4.w);

  const float zi = z[i];
  const float a0 = asc[2 * i], a1 = asc[2 * i + 1];
  const float na0 = fmaf(asc_decay[2 * i],     a0, zi * asc_amps[2 * i]);
  const float na1 = fmaf(asc_decay[2 * i + 1], a1, zi * asc_amps[2 * i + 1]);

  // input_current uses NEW psc sum + OLD asc sum (matches reference)
  const float icur = (npc.x + npc.y + npc.z + npc.w) + a0 + a1;

  const float vth = v_th[i];
  const float reset_c = zi * (v_reset[i] - vth);
  const float nv = fmaf(decay[i], v[i], cur_f[i] * (icur + e_l[i])) + reset_c;
  const float vsc = (nv - vth) / vth;

  float nz = (vsc > 0.0f) ? 1.0f : 0.0f;
  const float ro = r[i];
  if (ro > 0.0f) nz = 0.0f;                       // refractory gate
  const float nr = fmaxf(ro - 1.0f + nz * t_ref[i], 0.0f);

  *(float4*)(psc_rise + 4 * i) = npr;
  *(float4*)(psc + 4 * i) = npc;
  *(float4*)(rec_in + 4 * i) = make_float4(0.f, 0.f, 0.f, 0.f);  // reset accumulator
  asc[2 * i] = na0; asc[2 * i + 1] = na1;
  z[i] = nz; v[i] = nv; r[i] = nr;
  out_t[i] = nz;

  // spike bitmask for next step's scatter (NN is an exact multiple of 32)
  const unsigned m = ballot32(nz != 0.0f);
  if ((threadIdx.x & 31u) == 0u) z_bits[i >> 5] = m;
}

// ---------------------------------------------------------------------------
extern "C" void kernel_launch(void* const* d_in, const int* in_sizes, int n_in,
                              void* d_out, int out_size, void* d_ws, size_t ws_size,
                              hipStream_t stream) {
  const float* w_rec    = (const float*)d_in[0];
  const float* x_ext    = (const float*)d_in[1];
  const float* v0       = (const float*)d_in[2];
  const float* v_th     = (const float*)d_in[3];
  const float* v_reset  = (const float*)d_in[4];
  const float* t_ref    = (const float*)d_in[5];
  const float* decay    = (const float*)d_in[6];
  const float* cur_f    = (const float*)d_in[7];
  const float* e_l      = (const float*)d_in[8];
  const float* asc_amps = (const float*)d_in[9];
  const float* asc_dec  = (const float*)d_in[10];
  const float* syn_dec  = (const float*)d_in[11];
  const float* psc_ini  = (const float*)d_in[12];
  const int*   pre_idx  = (const int*)d_in[13];
  const int*   post_idx = (const int*)d_in[14];
  const int*   rcpt_idx = (const int*)d_in[15];
  float* out = (float*)d_out;

  // workspace carve-out (256B aligned slices)
  char* ws = (char*)d_ws;
  size_t off = 0;
  auto take = [&](size_t bytes) -> void* {
    off = (off + 255) & ~(size_t)255;
    void* p = ws + off;
    off += bytes;
    return p;
  };
  float*    z        = (float*)take(NN * sizeof(float));
  float*    v        = (float*)take(NN * sizeof(float));
  float*    r        = (float*)take(NN * sizeof(float));
  float*    asc      = (float*)take(2 * NN * sizeof(float));
  float*    psc      = (float*)take((size_t)NR * sizeof(float));
  float*    psc_rise = (float*)take((size_t)NR * sizeof(float));
  float*    rec_in   = (float*)take((size_t)NR * sizeof(float));
  unsigned* z_bits   = (unsigned*)take(ZWORDS_PAD * sizeof(unsigned));
  int*      slot     = (int*)take((size_t)EE * sizeof(int));
  (void)ws_size; (void)in_sizes; (void)n_in; (void)out_size;

  init_state_kernel<<<(NR + 255) / 256, 256, 0, stream>>>(
      v0, z, v, r, asc, psc, psc_rise, rec_in, z_bits);
  prep_slots_kernel<<<(EE + 255) / 256, 256, 0, stream>>>(
      post_idx, rcpt_idx, slot);

  const int scatter_blocks = 512;   // grid-stride over EE/4 quads
  const int neuron_blocks = (NN + 255) / 256;

  for (int t = 0; t < TT; ++t) {
    scatter_kernel<<<scatter_blocks, 256, 0, stream>>>(
        pre_idx, slot, w_rec, z_bits, rec_in);

    const float* x_t = x_ext + (size_t)t * NR;
    const float* x_n = (t + 1 < TT) ? (x_ext + (size_t)(t + 1) * NR) : nullptr;
    neuron_kernel<<<neuron_blocks, 256, 0, stream>>>(
        x_t, x_n, v_th, v_reset, t_ref, decay, cur_f, e_l,
        asc_amps, asc_dec, syn_dec, psc_ini,
        z, v, r, asc, psc, psc_rise, rec_in, z_bits,
        out + (size_t)t * NN);
  }
}